// GeneralisedSoftmax_72078141161977
// MI455X (gfx1250) — compile-verified
//
#include <hip/hip_runtime.h>
#include <math.h>

typedef float v2f __attribute__((ext_vector_type(2)));
typedef float v8f __attribute__((ext_vector_type(8)));

#define NN 2
#define LL 1024
#define SS 1024
#define HH 8
#define EE 64
#define KK 6
#define ROWS 16
#define KT_STRIDE 1033   // 1033 % 64 == 9 (coprime) -> conflict-free keyT access
#define QS 68            // 68 % 64 == 4 -> conflict-free A-fragment reads
#define NWAVES 8

struct SMem {
  float keyT[EE * KT_STRIDE];   // key transposed: keyT[e][s], 258 KB
  float qb[ROWS * QS];          // q block (16 x 64)
  float ht[ROWS * QS];          // htilde block (16 x 64), per code
  float k2[SS];                 // ||key_s||^2
  float kl[SS];                 // key_lengths_add[n][s]
  float q2[ROWS];               // ||htilde_r||^2
  float pi[ROWS][KK];           // mixture weights
  float redm[NWAVES][ROWS];     // cross-wave max partials
  float reds[NWAVES][ROWS];     // cross-wave sum partials
  float rowm[ROWS];
  float rowsum[ROWS];
};

static __device__ __forceinline__ v8f wmma_f32(v2f a, v2f b, v8f c) {
  // D = A(16x4,f32) * B(4x16,f32) + C(16x16,f32)
  return __builtin_amdgcn_wmma_f32_16x16x4_f32(false, a, false, b, (short)0, c,
                                               false, false);
}

static __device__ __forceinline__ float fast_tanh(float x) {
#if __has_builtin(__builtin_amdgcn_tanhf)
  return __builtin_amdgcn_tanhf(x);        // native v_tanh_f32 on gfx1250
#else
  // sign(x) * (1 - e^{-2|x|}) / (1 + e^{-2|x|}) : branch-free, no overflow
  float e2 = __expf(-2.0f * fabsf(x));
  return copysignf((1.0f - e2) / (1.0f + e2), x);
#endif
}

// Per-code logit transform over this wave's 8 tiles. CODE is compile-time so
// each specialization carries only its own handful of VALU ops per element.
template <int CODE>
static __device__ __forceinline__ void transform_tiles(
    v8f acc[8], const float* q2loc, float* vmax,
    int s0, int li, int hi, const float* __restrict__ maskb, const SMem& sm)
{
  #pragma unroll
  for (int st = 0; st < 8; ++st) {
    const int scol = s0 + st * 16 + li;
    const float k2v = sm.k2[scol];
    const float klv = sm.kl[scol];
    #pragma unroll
    for (int v = 0; v < 8; ++v) {
      const int r = v + hi * 8;
      const float d = acc[st][v];
      float x;
      if (CODE == 0) {                                   // L
        x = d;
      } else if (CODE == 3) {                            // Y: (d+1)^2
        float t = d + 1.0f; x = t * t;
      } else {
        float ns = q2loc[v] + k2v - 2.0f * d;
        if (CODE == 1)      x = -__logf(1.0f + ns);      // O
        else if (CODE == 2) x = -ns;                     // P
        else if (CODE == 4) x = __expf(-0.5f * ns);      // R
        else                x = __cosf(ns) * __expf(-ns);// W
      }
      x += klv + maskb[(size_t)r * SS + scol];
      acc[st][v] = x;
      vmax[v] = fmaxf(vmax[v], x);
    }
  }
}

__global__ __launch_bounds__(256)
void gsm_wmma_kernel(const float* __restrict__ q_g, const float* __restrict__ k_g,
                     const float* __restrict__ mask_g, const float* __restrict__ kl_g,
                     const float* __restrict__ M_g, const float* __restrict__ C_g,
                     float* __restrict__ out_g)
{
  __shared__ SMem sm;
  const int tid  = threadIdx.x;
  const int lane = tid & 31;
  const int wave = tid >> 5;
  const int li   = lane & 15;     // column / row-in-half index
  const int hi   = lane >> 4;     // half-wave select
  const int l0   = blockIdx.x * ROWS;
  const int h    = blockIdx.y;
  const int n    = blockIdx.z;
  const int s0   = wave * 128;    // this wave's 128-column strip

  // ---- stage q block, transposed key, key_lengths ----
  for (int idx = tid; idx < ROWS * EE; idx += 256) {
    int r = idx >> 6, e = idx & 63;
    sm.qb[r * QS + e] = q_g[(((size_t)n * LL + l0 + r) * HH + h) * EE + e];
  }
  for (int idx = tid; idx < EE * SS; idx += 256) {
    int e = idx & 63, s = idx >> 6;
    sm.keyT[e * KT_STRIDE + s] = k_g[(((size_t)n * SS + s) * HH + h) * EE + e];
  }
  for (int s = tid; s < SS; s += 256) sm.kl[s] = kl_g[n * SS + s];
  __syncthreads();

  // ---- k2[s] ----
  for (int s = tid; s < SS; s += 256) {
    float a2 = 0.f;
    #pragma unroll 8
    for (int e = 0; e < EE; ++e) { float v = sm.keyT[e * KT_STRIDE + s]; a2 += v * v; }
    sm.k2[s] = a2;
  }
  // ---- pi = softmax(q @ M[h]) over 6 codes ----
  if (tid < ROWS * KK) {
    int r = tid / KK, cc = tid % KK;
    float a = 0.f;
    for (int e = 0; e < EE; ++e)
      a += sm.qb[r * QS + e] * M_g[((size_t)h * EE + e) * KK + cc];
    sm.pi[r][cc] = a;
  }
  __syncthreads();
  if (tid < ROWS) {
    float mx = sm.pi[tid][0];
    for (int cc = 1; cc < KK; ++cc) mx = fmaxf(mx, sm.pi[tid][cc]);
    float den = 0.f; float ex[KK];
    for (int cc = 0; cc < KK; ++cc) { ex[cc] = __expf(sm.pi[tid][cc] - mx); den += ex[cc]; }
    for (int cc = 0; cc < KK; ++cc) sm.pi[tid][cc] = ex[cc] / den;
  }

  v8f outacc[8];
  #pragma unroll
  for (int st = 0; st < 8; ++st) outacc[st] = v8f{};

  const float* maskb = mask_g + ((size_t)n * LL + l0) * SS;

  for (int c = 0; c < KK; ++c) {
    __syncthreads();   // ht / q2 / reduction buffers free for reuse

    // ---- htilde = tanh(qb @ C[c,h]) : waves 0..3 each compute one 16-col tile ----
    if (wave < 4) {
      const float* Cb = C_g + ((size_t)c * HH + h) * EE * EE;
      const int f0 = wave * 16;
      v8f hacc = v8f{};
      #pragma unroll
      for (int kc = 0; kc < 16; ++kc) {
        const int kb = kc * 4 + hi * 2;
        v2f a, b;
        a.x = sm.qb[li * QS + kb];
        a.y = sm.qb[li * QS + kb + 1];
        b.x = Cb[(size_t)kb * EE + f0 + li];
        b.y = Cb[(size_t)(kb + 1) * EE + f0 + li];
        hacc = wmma_f32(a, b, hacc);
      }
      #pragma unroll
      for (int v = 0; v < 8; ++v)
        sm.ht[(v + hi * 8) * QS + f0 + li] = fast_tanh(hacc[v]);
    }
    __syncthreads();
    if (tid < ROWS) {
      float a2 = 0.f;
      for (int e = 0; e < EE; ++e) { float v = sm.ht[tid * QS + e]; a2 += v * v; }
      sm.q2[tid] = a2;
    }
    __syncthreads();

    // ---- logits = htilde @ key^T over this wave's 128-column strip ----
    v8f acc[8];
    #pragma unroll
    for (int st = 0; st < 8; ++st) acc[st] = v8f{};
    #pragma unroll
    for (int kc = 0; kc < 16; ++kc) {
      const int kb = kc * 4 + hi * 2;
      v2f a;
      a.x = sm.ht[li * QS + kb];
      a.y = sm.ht[li * QS + kb + 1];
      #pragma unroll
      for (int st = 0; st < 8; ++st) {
        v2f b;
        b.x = sm.keyT[kb * KT_STRIDE + s0 + st * 16 + li];
        b.y = sm.keyT[(kb + 1) * KT_STRIDE + s0 + st * 16 + li];
        acc[st] = wmma_f32(a, b, acc[st]);
      }
    }

    // ---- kernel transform + additive mask; track per-row max ----
    float q2loc[8], vmax[8];
    #pragma unroll
    for (int v = 0; v < 8; ++v) { q2loc[v] = sm.q2[v + hi * 8]; vmax[v] = -3.4e38f; }
    switch (c) {
      case 0: transform_tiles<0>(acc, q2loc, vmax, s0, li, hi, maskb, sm); break;
      case 1: transform_tiles<1>(acc, q2loc, vmax, s0, li, hi, maskb, sm); break;
      case 2: transform_tiles<2>(acc, q2loc, vmax, s0, li, hi, maskb, sm); break;
      case 3: transform_tiles<3>(acc, q2loc, vmax, s0, li, hi, maskb, sm); break;
      case 4: transform_tiles<4>(acc, q2loc, vmax, s0, li, hi, maskb, sm); break;
      default: transform_tiles<5>(acc, q2loc, vmax, s0, li, hi, maskb, sm); break;
    }

    // ---- row max: xor-shuffle within 16-lane half, then across 8 waves ----
    #pragma unroll
    for (int m = 1; m <= 8; m <<= 1) {
      #pragma unroll
      for (int v = 0; v < 8; ++v)
        vmax[v] = fmaxf(vmax[v], __shfl_xor(vmax[v], m, 32));
    }
    if (li == 0) {
      #pragma unroll
      for (int v = 0; v < 8; ++v) sm.redm[wave][v + hi * 8] = vmax[v];
    }
    __syncthreads();
    if (tid < ROWS) {
      float mx = sm.redm[0][tid];
      for (int w = 1; w < NWAVES; ++w) mx = fmaxf(mx, sm.redm[w][tid]);
      sm.rowm[tid] = mx;
    }
    __syncthreads();

    // ---- exp and row sum ----
    float rmax[8], vsum[8];
    #pragma unroll
    for (int v = 0; v < 8; ++v) { rmax[v] = sm.rowm[v + hi * 8]; vsum[v] = 0.f; }
    #pragma unroll
    for (int st = 0; st < 8; ++st) {
      #pragma unroll
      for (int v = 0; v < 8; ++v) {
        float e = __expf(acc[st][v] - rmax[v]);
        acc[st][v] = e;
        vsum[v] += e;
      }
    }
    #pragma unroll
    for (int m = 1; m <= 8; m <<= 1) {
      #pragma unroll
      for (int v = 0; v < 8; ++v)
        vsum[v] += __shfl_xor(vsum[v], m, 32);
    }
    if (li == 0) {
      #pragma unroll
      for (int v = 0; v < 8; ++v) sm.reds[wave][v + hi * 8] = vsum[v];
    }
    __syncthreads();
    if (tid < ROWS) {
      float s = 0.f;
      for (int w = 0; w < NWAVES; ++w) s += sm.reds[w][tid];
      sm.rowsum[tid] = s;
    }
    __syncthreads();

    // ---- accumulate pi-weighted softmax into output tiles ----
    #pragma unroll
    for (int v = 0; v < 8; ++v) {
      const float wgt = sm.pi[v + hi * 8][c] / sm.rowsum[v + hi * 8];
      #pragma unroll
      for (int st = 0; st < 8; ++st)
        outacc[st][v] += acc[st][v] * wgt;
    }
  }

  // ---- store out[n,h,l,s], written exactly once ----
  float* ob = out_g + (((size_t)n * HH + h) * LL + l0) * SS;
  #pragma unroll
  for (int st = 0; st < 8; ++st) {
    #pragma unroll
    for (int v = 0; v < 8; ++v)
      ob[(size_t)(v + hi * 8) * SS + s0 + st * 16 + li] = outacc[st][v];
  }
}

extern "C" void kernel_launch(void* const* d_in, const int* in_sizes, int n_in,
                              void* d_out, int out_size, void* d_ws, size_t ws_size,
                              hipStream_t stream) {
  (void)in_sizes; (void)n_in; (void)out_size; (void)d_ws; (void)ws_size;
  const float* q    = (const float*)d_in[0];
  const float* k    = (const float*)d_in[1];
  const float* mask = (const float*)d_in[2];
  const float* kl   = (const float*)d_in[3];
  const float* M    = (const float*)d_in[4];
  const float* C    = (const float*)d_in[5];
  float* out = (float*)d_out;
  dim3 grid(LL / ROWS, HH, NN);   // (64, 8, 2) workgroups, 8 waves each
  gsm_wmma_kernel<<<grid, dim3(256), 0, stream>>>(q, k, mask, kl, M, C, out);
}